// MeanAggregator_56831007261162
// MI455X (gfx1250) — compile-verified
//
#include <hip/hip_runtime.h>

// MeanAggregator (GraphSAGE mean) for MI455X / gfx1250.
//
// out[b, d] = (1/S) * sum_s features[neigh_idx[b, s], d]
//   B = 50000, S = 10, N = 1e6, D = 128, fp32.
//
// Memory-bound: ~256MB gather + 25.6MB store => ~12us floor at 23.3 TB/s.
// Sustaining that needs ~260 outstanding 128B loads per SIMD (~1us latency),
// i.e. >= ~7 waves/SIMD each with a ~20-40 deep load pipeline. So the top
// priority is VGPR count (occupancy), then load-issue efficiency.
//
// Reduction expressed as D = A x B with V_WMMA_F32_16X16X4_F32:
//   per group of 16 nodes, A(16x160) is a constant block-diagonal 1/S
//   selection matrix (chunked K=4), B(160x16) is the gathered feature slice.
// fp32 WMMA accumulate -> numerically equivalent to the reference.
//
// v3 changes (from disasm feedback):
//  - 256-thread block, one 16-wide column tile per wave (no t-loop): halves
//    the static pipelined code and live ranges.
//  - Loads index the ORIGINAL uniform kernarg pointer with one u32 offset
//    (uniform_base + zext(u32)) - the canonical GlobalSAddr fold pattern -
//    instead of a pre-offset pointer, to get saddr-form global_load_b32 and
//    drop the per-chunk v_add_nc_u64 pairs.
//  - #pragma unroll 20: keeps a deep load->wmma pipeline but halves the
//    offset/data register footprint, targeting <256 VGPRs (no s_set_vgpr_msb
//    churn) and >=6 waves/SIMD.

typedef float v2f __attribute__((ext_vector_type(2)));
typedef float v8f __attribute__((ext_vector_type(8)));

#define SAMP   10                   // S: neighbors per node
#define FDIM   128                  // D: feature dim
#define NPG    16                   // nodes per group (WMMA M)
#define SLOTS  (NPG * SAMP)         // 160 sample slots per group (WMMA K total)
#define KCH    (SLOTS / 4)          // 40 K-chunks of 4 (V_WMMA_F32_16X16X4_F32)
#define WAVES  8                    // one column tile per wave (D/16 = 8)
#define BLOCK  (WAVES * 32)
#define RCPS   (1.0f / (float)SAMP)

__global__ __launch_bounds__(BLOCK)
void meanagg_wmma_kernel(const int* __restrict__ neigh_idx,
                         const float* __restrict__ features,
                         float* __restrict__ out,
                         int nBatch) {
    // Row byte-offsets of the 160 gathered feature rows for this group.
    __shared__ __align__(16) unsigned rowbytes[SLOTS];

    const int group = blockIdx.x;
    const int node0 = group * NPG;
    const int tid   = threadIdx.x;

    // --- Stage 1: indices -> byte offsets (coalesced, 160 x u32 in LDS) ----
    for (int s = tid; s < SLOTS; s += BLOCK) {
        const int node = node0 + s / SAMP;
        unsigned idx = 0u;
        if (node < nBatch) idx = (unsigned)neigh_idx[node0 * SAMP + s];
        rowbytes[s] = idx * (unsigned)(FDIM * 4);   // < 2^29, fits u32
    }
    __syncthreads();

    // --- Stage 2: WMMA reduction ------------------------------------------
    const int lane = tid & 31;
    const int wave = tid >> 5;             // = column tile ct (0..7)
    const int h    = lane >> 4;            // half-wave: K pair / M bank
    const int c    = lane & 15;            // N column (B/D) and M row (A)
    const int mS   = c * SAMP;             // per-lane slot range base for A

    // Per-lane column byte offset within a feature row: tile*64 + c*4.
    const unsigned cb = (unsigned)(wave * 64 + c * 4);

    const char* fbase = (const char*)features;

    v8f acc = {0.f, 0.f, 0.f, 0.f, 0.f, 0.f, 0.f, 0.f};

#pragma unroll 20
    for (int kc = 0; kc < KCH; ++kc) {
        const int k0 = kc * 4;
        const int ka = k0 + 2 * h;         // this half-wave's K pair

        // Two consecutive row offsets -> one ds_load_b64 (ka is even).
        const uint2 off2 = *reinterpret_cast<const uint2*>(&rowbytes[ka]);

        // Single u32 voffset per load: uniform base + zext(u32) -> saddr form.
        unsigned vx = off2.x + cb;
        unsigned vy = off2.y + cb;
        __builtin_assume(vx < 0x40000000u);
        __builtin_assume(vy < 0x40000000u);

        // B operand (4x16 fp32, 2 VGPRs):
        //   b.x: row K=ka, col ct*16+c    b.y: row K=ka+1, col ct*16+c
        // Half-waves coalesce into 64B segments; the 8 column tiles of a row
        // share L2 lines, so HBM reads each 512B row once.
        v2f b;
        b.x = *(const float*)(fbase + vx);
        b.y = *(const float*)(fbase + vy);

        // A operand (16x4 fp32, 2 VGPRs): constant block-diagonal 1/S.
        //   A[m,k] = 1/S  iff  slot k belongs to node m (k in [mS, mS+S))
        v2f a;
        a.x = ((unsigned)(ka     - mS) < (unsigned)SAMP) ? RCPS : 0.0f;
        a.y = ((unsigned)(ka + 1 - mS) < (unsigned)SAMP) ? RCPS : 0.0f;

        // D(16x16) += A(16x4) x B(4x16), fp32 accumulate.
        acc = __builtin_amdgcn_wmma_f32_16x16x4_f32(
            /*neg_a=*/false, a, /*neg_b=*/false, b,
            /*c_mod=*/(short)0, acc,
            /*reuse_a=*/false, /*reuse_b=*/false);
    }

    // C/D layout: VGPR r -> M = r + 8*h, N = c.  This wave's tile = wave.
#pragma unroll
    for (int r = 0; r < 8; ++r) {
        const int node = node0 + r + 8 * h;
        if (node < nBatch) out[node * FDIM + wave * 16 + c] = acc[r];
    }
}

extern "C" void kernel_launch(void* const* d_in, const int* in_sizes, int n_in,
                              void* d_out, int out_size, void* d_ws, size_t ws_size,
                              hipStream_t stream) {
    const int*   neigh_idx = (const int*)d_in[0];    // [B, S] int32
    const float* features  = (const float*)d_in[1];  // [N, D] fp32
    float*       out       = (float*)d_out;          // [B, D] fp32

    const int nBS    = in_sizes[0];          // B * S
    const int nBatch = nBS / SAMP;           // B
    const int groups = (nBatch + NPG - 1) / NPG;

    meanagg_wmma_kernel<<<groups, BLOCK, 0, stream>>>(
        neigh_idx, features, out, nBatch);
}